// RadialDescriptor_14869176778800
// MI455X (gfx1250) — compile-verified
//
#include <hip/hip_runtime.h>

// ---------------------------------------------------------------------------
// RadialDescriptor: g[e,d] = sum_k c_table[ti,tj,d,k] * f[e,k]
//   f[e,k] = (T_k(x)+1) * 0.5 * fc(r),  x = 2*(r/rc-1)^2-1
// Restructured: g[e,d] = sum_j (fc * x^j) * W[pair][j][d]
//   W[pair][j][d] = 0.5*( sum_k c[pair][d][k]*CHEB[k][j] + (j==0)*sum_k c[pair][d][k] )
// Edges are bucketed by (type_i,type_j) pair inside each block, then
// contracted with V_WMMA_F32_16X16X4_F32 (two per 16-edge tile, K=8 total,
// exact f32 math). fc is folded into the A rows so the WMMA output is final.
// Records hold only {x, fc} + output byte offset; A monomials are rebuilt
// per tile in VALU (co-executes with the XDL matrix pipe).
// ---------------------------------------------------------------------------

typedef float v8f __attribute__((ext_vector_type(8)));
typedef float v2f __attribute__((ext_vector_type(2)));

#define BLOCK_E   2048     // edges per block
#define NTHREADS  256
#define EPT       8        // edges per thread in phase 1
#define NPAIR     64
#define RECCAP    2304     // 2048 + 64*3 pad + margin
#define RC        5.0f
#define PI_F      3.14159265358979323846f

// Chebyshev T_k monomial coefficients, CHEB[k*8+j] = coeff of x^j in T_k
__device__ __constant__ float CHEB[64] = {
     1.f,  0.f,   0.f,   0.f,   0.f,    0.f,  0.f,  0.f,   // T0
     0.f,  1.f,   0.f,   0.f,   0.f,    0.f,  0.f,  0.f,   // T1
    -1.f,  0.f,   2.f,   0.f,   0.f,    0.f,  0.f,  0.f,   // T2
     0.f, -3.f,   0.f,   4.f,   0.f,    0.f,  0.f,  0.f,   // T3
     1.f,  0.f,  -8.f,   0.f,   8.f,    0.f,  0.f,  0.f,   // T4
     0.f,  5.f,   0.f, -20.f,   0.f,   16.f,  0.f,  0.f,   // T5
    -1.f,  0.f,  18.f,   0.f, -48.f,    0.f, 32.f,  0.f,   // T6
     0.f, -7.f,   0.f,  56.f,   0.f, -112.f,  0.f, 64.f    // T7
};

__global__ __launch_bounds__(NTHREADS) void radial_descriptor_kernel(
    const float* __restrict__ r_ij,
    const int*   __restrict__ type_i,
    const int*   __restrict__ type_j,
    const float* __restrict__ c_table,   // [8][8][8][8] = pair*64 + d*8 + k
    float*       __restrict__ out,       // [E][8]
    int E)
{
    __shared__ __align__(16) float    Wlds[NPAIR * 64];   // 16 KB
    __shared__ __align__(8)  float2   xfc[RECCAP];        // 18 KB {x, fc}
    __shared__ __align__(16) unsigned eoff[RECCAP];       // 9 KB  (e*32 bytes)
    __shared__ unsigned counts[NPAIR];
    __shared__ unsigned starts[NPAIR];
    __shared__ unsigned offs[NPAIR];
    __shared__ unsigned scanbuf[NPAIR];

    const int tid  = threadIdx.x;
    const int base = blockIdx.x * BLOCK_E;

    // ---- Build W[pair][j][d] in LDS from the 16 KB (L2-resident) c_table ----
    for (int idx = tid; idx < NPAIR * 64; idx += NTHREADS) {
        const int p = idx >> 6;
        const int j = (idx >> 3) & 7;
        const int d = idx & 7;
        const float* c = c_table + (p << 6) + (d << 3);   // c[p][d][k]
        float acc = 0.f, s = 0.f;
        #pragma unroll
        for (int k = 0; k < 8; ++k) {
            acc += CHEB[k * 8 + j] * c[k];
            s   += c[k];
        }
        Wlds[idx] = 0.5f * (acc + (j == 0 ? s : 0.f));
    }
    if (tid < NPAIR) counts[tid] = 0u;
    __syncthreads();

    // ---- Phase 1: load edges, compute x & fc, count buckets ----
    float xv[EPT], fcv[EPT];
    int   pv[EPT];
    const bool full = (base + BLOCK_E <= E);
    if (full) {
        const float4* rp  = (const float4*)(r_ij  + base) + 2 * tid;
        const int4*   tip = (const int4*)(type_i + base) + 2 * tid;
        const int4*   tjp = (const int4*)(type_j + base) + 2 * tid;
        const float4 rq0 = rp[0],  rq1 = rp[1];
        const int4   ti0 = tip[0], ti1 = tip[1];
        const int4   tj0 = tjp[0], tj1 = tjp[1];
        const float rr[EPT] = {rq0.x, rq0.y, rq0.z, rq0.w,
                               rq1.x, rq1.y, rq1.z, rq1.w};
        const int   ti[EPT] = {ti0.x, ti0.y, ti0.z, ti0.w,
                               ti1.x, ti1.y, ti1.z, ti1.w};
        const int   tj[EPT] = {tj0.x, tj0.y, tj0.z, tj0.w,
                               tj1.x, tj1.y, tj1.z, tj1.w};
        #pragma unroll
        for (int i = 0; i < EPT; ++i) {
            const float r = rr[i];
            fcv[i] = (r < RC) ? (0.5f * __cosf((PI_F / RC) * r) + 0.5f) : 0.f;
            const float t = r * (1.0f / RC) - 1.0f;
            xv[i] = 2.0f * t * t - 1.0f;
            pv[i] = ((ti[i] & 7) << 3) | (tj[i] & 7);
            atomicAdd(&counts[pv[i]], 1u);
        }
    } else {
        #pragma unroll
        for (int i = 0; i < EPT; ++i) {
            const int e = base + tid * EPT + i;
            if (e < E) {
                const float r = r_ij[e];
                fcv[i] = (r < RC) ? (0.5f * __cosf((PI_F / RC) * r) + 0.5f) : 0.f;
                const float t = r * (1.0f / RC) - 1.0f;
                xv[i] = 2.0f * t * t - 1.0f;
                pv[i] = ((type_i[e] & 7) << 3) | (type_j[e] & 7);
                atomicAdd(&counts[pv[i]], 1u);
            } else {
                xv[i] = 0.f; fcv[i] = 0.f; pv[i] = -1;
            }
        }
    }
    __syncthreads();

    // ---- Exclusive scan of pad4(count) to get 4-aligned bucket starts ----
    if (tid < NPAIR) scanbuf[tid] = (counts[tid] + 3u) & ~3u;
    __syncthreads();
    for (int off = 1; off < NPAIR; off <<= 1) {
        unsigned v = 0;
        if (tid < NPAIR && tid >= off) v = scanbuf[tid - off];
        __syncthreads();
        if (tid < NPAIR) scanbuf[tid] += v;
        __syncthreads();
    }
    if (tid < NPAIR) {
        const unsigned st = scanbuf[tid] - ((counts[tid] + 3u) & ~3u);
        starts[tid] = st;
        offs[tid]   = st;
    }
    __syncthreads();

    // ---- Phase 1b: place records: {x, fc} + output byte offset ----
    #pragma unroll
    for (int i = 0; i < EPT; ++i) {
        const int p = pv[i];
        if (p >= 0) {
            const unsigned slot = atomicAdd(&offs[p], 1u);
            xfc[slot]  = make_float2(xv[i], fcv[i]);
            eoff[slot] = ((unsigned)(base + tid * EPT + i)) * 32u; // e*8 floats
        }
    }
    __syncthreads();

    // ---- Phase 2: per-pair GEMM tiles via V_WMMA_F32_16X16X4_F32 ----
    const int lane    = tid & 31;
    const int halfsel = lane >> 4;          // K-block selector (ISA A/B layout)
    const int m16     = lane & 15;          // A row / D column
    const int k0      = 2 * halfsel;
    const bool hs     = (halfsel != 0);
    const unsigned fourm = 4u * (unsigned)m16;
    char* outc = (char*)out;

    for (int p = (tid >> 5); p < NPAIR; p += (NTHREADS / 32)) {
        const unsigned cnt = (unsigned)__builtin_amdgcn_readfirstlane((int)counts[p]);
        if (cnt == 0u) continue;
        const unsigned st = (unsigned)__builtin_amdgcn_readfirstlane((int)starts[p]);

        // B operands (shared across all tiles of this pair).
        // B[v][lane] = W[K = k0+v (+4 for 2nd wmma)][N = lane%16]
        v2f B0 = {0.f, 0.f}, B1 = {0.f, 0.f};
        if (m16 < 8) {
            const float* wp = &Wlds[p * 64];
            B0[0] = wp[(k0 + 0) * 8 + m16];
            B0[1] = wp[(k0 + 1) * 8 + m16];
            B1[0] = wp[(k0 + 4) * 8 + m16];
            B1[1] = wp[(k0 + 5) * 8 + m16];
        }

        const unsigned ntiles = (cnt + 15u) >> 4;
        for (unsigned t = 0; t < ntiles; ++t) {
            const unsigned t0 = t << 4;

            // Per-lane A row: rebuild fc*x^j from {x, fc} (unclamped read;
            // stale rows only affect their own masked-out D rows).
            const float2 xf = xfc[st + t0 + (unsigned)m16];
            const float x = xf.x, fc = xf.y;
            const float x2 = x * x,  x3 = x2 * x,  x4 = x2 * x2;
            const float x5 = x4 * x, x6 = x4 * x2, x7 = x4 * x3;
            const v2f A0 = {fc * (hs ? x2 : 1.0f), fc * (hs ? x3 : x)};
            const v2f A1 = {fc * (hs ? x6 : x4),   fc * (hs ? x7 : x5)};

            // Output row byte-offsets for this lane's 8 D rows.
            const unsigned rbase = st + t0 + 8u * (unsigned)halfsel;
            const uint4 ea = *(const uint4*)&eoff[rbase];
            const uint4 eb = *(const uint4*)&eoff[rbase + 4];
            const unsigned eo[8] = {ea.x, ea.y, ea.z, ea.w,
                                    eb.x, eb.y, eb.z, eb.w};

            v8f acc = {};
            acc = __builtin_amdgcn_wmma_f32_16x16x4_f32(
                      false, A0, false, B0, (short)0, acc, false, false);
            acc = __builtin_amdgcn_wmma_f32_16x16x4_f32(
                      false, A1, false, B1, (short)0, acc, false, false);

            // D readout: lane holds D[M = v + 8*halfsel][N = m16].
            if (t0 + 16u <= cnt) {            // full tile (scalar branch)
                if (m16 < 8) {
                    #pragma unroll
                    for (int v = 0; v < 8; ++v) {
                        *(float*)(outc + (eo[v] + fourm)) = acc[v];
                    }
                }
            } else {                           // partial tail tile
                const unsigned nv    = cnt - t0;
                const unsigned limit = (m16 < 8) ? nv : 0u;
                const unsigned Mbase = hs ? 8u : 0u;
                #pragma unroll
                for (int v = 0; v < 8; ++v) {
                    if (Mbase + (unsigned)v < limit) {
                        *(float*)(outc + (eo[v] + fourm)) = acc[v];
                    }
                }
            }
        }
    }
}

extern "C" void kernel_launch(void* const* d_in, const int* in_sizes, int n_in,
                              void* d_out, int out_size, void* d_ws, size_t ws_size,
                              hipStream_t stream) {
    const float* r_ij    = (const float*)d_in[0];
    const int*   type_i  = (const int*)d_in[1];
    const int*   type_j  = (const int*)d_in[2];
    const float* c_table = (const float*)d_in[3];
    float*       out     = (float*)d_out;
    const int E = in_sizes[0];

    const int nblocks = (E + BLOCK_E - 1) / BLOCK_E;
    radial_descriptor_kernel<<<nblocks, NTHREADS, 0, stream>>>(
        r_ij, type_i, type_j, c_table, out, E);
}